// Transformer_3D_12163347382622
// MI455X (gfx1250) — compile-verified
//
#include <hip/hip_runtime.h>

// ---------------- problem constants ----------------
#define L_   2
#define B_   2
#define C_   256
#define H_   8
#define HD_  32
#define T_   16
#define S_   512
#define DFF_ 1024
#define TS_  (T_ * S_)                 // 8192
#define NTOT_ ((size_t)B_ * C_ * TS_)  // 4,194,304 floats = 16 MB

typedef __attribute__((ext_vector_type(2))) float v2f;
typedef __attribute__((ext_vector_type(8))) float v8f;

// CDNA5 fp32 matrix core: D(16x16) = A(16x4) * B(4x16) + C, f32 throughout.
__device__ __forceinline__ v8f wmma4(v2f a, v2f b, v8f c) {
  return __builtin_amdgcn_wmma_f32_16x16x4_f32(
      /*neg_a=*/false, a, /*neg_b=*/false, b,
      /*c_mod=*/(short)0, c, /*reuse_a=*/false, /*reuse_b=*/false);
}

// CDNA5 async global->LDS copy (ASYNCcnt-tracked, no VGPR round trip).
// VDST VGPR carries the LDS byte offset == low 32 bits of the generic pointer
// (shared aperture keeps the LDS offset in addr[31:0]).
__device__ __forceinline__ void async_g2l_b32(void* lds, const void* gptr) {
  const unsigned int loff = (unsigned int)(unsigned long long)lds;
  asm volatile("global_load_async_to_lds_b32 %0, %1, off"
               :: "v"(loff), "v"(gptr) : "memory");
}
__device__ __forceinline__ void wait_async0() {
  asm volatile("s_wait_asynccnt 0" ::: "memory");
}

// ---------------- LayerNorm over channel dim of [B,C,T,S] ----------------
__global__ __launch_bounds__(C_)
void ln_kernel(const float* __restrict__ X, float* __restrict__ Y,
               const float* __restrict__ g, const float* __restrict__ bt) {
  __shared__ float red[C_];
  const int p  = blockIdx.x;            // b*TS + ts
  const int b  = p / TS_;
  const int ts = p - b * TS_;
  const int c  = threadIdx.x;
  const size_t idx = ((size_t)b * C_ + c) * TS_ + ts;
  const float x = X[idx];

  red[c] = x; __syncthreads();
  for (int s = C_ / 2; s > 0; s >>= 1) {
    if (c < s) red[c] += red[c + s];
    __syncthreads();
  }
  const float mean = red[0] * (1.0f / C_);
  __syncthreads();
  const float d = x - mean;
  red[c] = d * d; __syncthreads();
  for (int s = C_ / 2; s > 0; s >>= 1) {
    if (c < s) red[c] += red[c + s];
    __syncthreads();
  }
  const float var = red[0] * (1.0f / C_);
  Y[idx] = d * rsqrtf(var + 1e-5f) * g[c] + bt[c];
}

// ---------------- WMMA GEMM ----------------
// Y[b,m,n] = W[m,:]·(X1+X2)[b,:,n] (+bias, +res, relu)
// X element (b,k,n) -> X + (b*KDIM + k)*IN_STRIDE + inOff + n
// Y element (b,m,n) -> Y + (b*M + m)*outStride + outOff + n
// Block: 128 thr = 4 waves, 64x64 tile; wave: 32x32 (2x2 WMMA tiles).
// Compile-time KDIM/IN_STRIDE keep all companion loads at immediate offsets.
template <int KDIM, int IN_STRIDE, bool HAS_X2, bool HAS_RES, bool RELU>
__global__ __launch_bounds__(128)
void gemm_kernel(const float* __restrict__ W, const float* __restrict__ bias,
                 const float* __restrict__ X1, const float* __restrict__ X2,
                 const float* __restrict__ Res, float* __restrict__ Y,
                 int M, int inOff, int outStride, int outOff) {
  const int b     = blockIdx.z;
  const int nBase = blockIdx.x * 64;
  const int mBase = blockIdx.y * 64;
  const int w     = threadIdx.x >> 5;
  const int lane  = threadIdx.x & 31;
  const int lh    = lane >> 4;          // lane half (0/1)
  const int lm    = lane & 15;
  const int wm    = mBase + ((w >> 1) << 5);
  const int wn    = nBase + ((w & 1) << 5);

  const float* wp = W + (size_t)(wm + lm) * KDIM + 2 * lh;
  const float* xp = X1 + ((size_t)b * KDIM + 2 * lh) * IN_STRIDE + inOff + wn + lm;
  const float* xq = nullptr;
  if constexpr (HAS_X2)
    xq = X2 + ((size_t)b * KDIM + 2 * lh) * IN_STRIDE + inOff + wn + lm;

  const v8f vz = {0, 0, 0, 0, 0, 0, 0, 0};
  v8f acc[2][2];
  acc[0][0] = vz; acc[0][1] = vz; acc[1][0] = vz; acc[1][1] = vz;

  #pragma unroll 4
  for (int kk = 0; kk < KDIM; kk += 4) {
    const float2 wa0 = *(const float2*)(wp);
    const float2 wa1 = *(const float2*)(wp + 16 * KDIM);
    v2f a0; a0.x = wa0.x; a0.y = wa0.y;
    v2f a1; a1.x = wa1.x; a1.y = wa1.y;

    float b00 = xp[0];
    float b01 = xp[IN_STRIDE];
    float b10 = xp[16];
    float b11 = xp[IN_STRIDE + 16];
    if constexpr (HAS_X2) {            // fused "+ pos"
      b00 += xq[0];
      b01 += xq[IN_STRIDE];
      b10 += xq[16];
      b11 += xq[IN_STRIDE + 16];
    }
    v2f bb0; bb0.x = b00; bb0.y = b01;
    v2f bb1; bb1.x = b10; bb1.y = b11;

    acc[0][0] = wmma4(a0, bb0, acc[0][0]);
    acc[0][1] = wmma4(a0, bb1, acc[0][1]);
    acc[1][0] = wmma4(a1, bb0, acc[1][0]);
    acc[1][1] = wmma4(a1, bb1, acc[1][1]);

    wp += 4;
    xp += (size_t)4 * IN_STRIDE;
    if constexpr (HAS_X2) xq += (size_t)4 * IN_STRIDE;
  }

  #pragma unroll
  for (int mt = 0; mt < 2; ++mt) {
    #pragma unroll
    for (int nt = 0; nt < 2; ++nt) {
      const int col = wn + nt * 16 + lm;
      #pragma unroll
      for (int j = 0; j < 8; ++j) {
        const int row = wm + mt * 16 + j + 8 * lh;
        float val = acc[mt][nt][j] + bias[row];
        const size_t oidx = ((size_t)b * M + row) * outStride + outOff + col;
        if constexpr (HAS_RES) val += Res[oidx];
        if constexpr (RELU) val = fmaxf(val, 0.0f);
        Y[oidx] = val;
      }
    }
  }
}

// ---------------- distance-gated masked attention ----------------
// grid: (S/16 query tiles, H*T, B), block 64 thr = 2 waves.
#define SROW 520
__global__ __launch_bounds__(64)
void attn_kernel(const float* __restrict__ q, const float* __restrict__ k,
                 const float* __restrict__ v, const float* __restrict__ coord,
                 const unsigned char* __restrict__ mask,
                 float* __restrict__ vals) {
  __shared__ float sc[16][SROW];
  __shared__ float kc[S_ * 3];          // interleaved xyz for all 512 positions
  __shared__ unsigned int msk4[S_ / 4]; // raw mask bytes, 4 per dword
  __shared__ float red[4][16];
  __shared__ float rowmax[16], rowsum[16];

  const int qBase = blockIdx.x * 16;
  const int h     = blockIdx.y / T_;
  const int t     = blockIdx.y % T_;
  const int b     = blockIdx.z;
  const int tid   = threadIdx.x;
  const int w     = tid >> 5;
  const int lane  = tid & 31;
  const int lh    = lane >> 4;
  const int lm    = lane & 15;

  const size_t head_off = ((size_t)(b * C_ + h * HD_) * T_ + t) * S_;
  const float* qp = q + head_off;
  const float* kp = k + head_off;
  const float* vp = v + head_off;

  // async-stage coords (6 KB) + mask (512 B) straight into LDS
  const size_t cbase = (size_t)(b * T_ + t) * S_;
  {
    const float* cg = coord + cbase * 3;
    for (int i = tid; i < S_ * 3; i += 64) async_g2l_b32(&kc[i], cg + i);
    const unsigned int* mg = (const unsigned int*)(mask + cbase);
    for (int i = tid; i < S_ / 4; i += 64) async_g2l_b32(&msk4[i], mg + i);
    wait_async0();
  }
  __syncthreads();

  // phase 1: scores = (Q·K^T)/sqrt(HD) * gate, masked -> LDS
  const v8f vz = {0, 0, 0, 0, 0, 0, 0, 0};
  const float scale = 0.17677669529663687f;  // 1/sqrt(32)
  const unsigned char* mskb = (const unsigned char*)msk4;

  // query coords for this lane's 8 output rows, hoisted to registers
  float qx[8], qy[8], qz[8];
  #pragma unroll
  for (int j = 0; j < 8; ++j) {
    const int qi3 = (qBase + j + 8 * lh) * 3;
    qx[j] = kc[qi3 + 0]; qy[j] = kc[qi3 + 1]; qz[j] = kc[qi3 + 2];
  }

  const float* qcol = qp + (size_t)(2 * lh) * TS_ + qBase + lm;  // A frag base
  for (int i = 0; i < 16; ++i) {
    const int n0 = (w * 16 + i) * 16;
    const float* kcol = kp + (size_t)(2 * lh) * TS_ + n0 + lm;   // B frag base
    v8f acc = vz;
    #pragma unroll
    for (int kk = 0; kk < HD_; kk += 4) {
      v2f a;  a.x  = qcol[(size_t)kk * TS_];
              a.y  = qcol[(size_t)(kk + 1) * TS_];
      v2f bb; bb.x = kcol[(size_t)kk * TS_];
              bb.y = kcol[(size_t)(kk + 1) * TS_];
      acc = wmma4(a, bb, acc);
    }
    const int col = n0 + lm;
    const float kx = kc[col * 3 + 0];
    const float ky = kc[col * 3 + 1];
    const float kz = kc[col * 3 + 2];
    const bool  mk = mskb[col] != 0;
    #pragma unroll
    for (int j = 0; j < 8; ++j) {
      const int row = j + 8 * lh;
      float val = acc[j] * scale;
      const float dx = qx[j] - kx;
      const float dy = qy[j] - ky;
      const float dz = qz[j] - kz;
      if (dx * dx + dy * dy + dz * dz >= 625.0f) val = 0.0f;  // gate
      if (mk) val = -9e15f;                                   // key padding
      sc[row][col] = val;
    }
  }
  __syncthreads();

  // phase 2: softmax over 512 keys per row (two-pass, 4 partials/row)
  const int row  = tid & 15;
  const int part = tid >> 4;
  {
    float m = -3.0e38f;
    for (int ci = 0; ci < 128; ++ci) m = fmaxf(m, sc[row][part * 128 + ci]);
    red[part][row] = m;
  }
  __syncthreads();
  if (part == 0)
    rowmax[row] = fmaxf(fmaxf(red[0][row], red[1][row]),
                        fmaxf(red[2][row], red[3][row]));
  __syncthreads();
  {
    const float rm = rowmax[row];
    float s = 0.0f;
    for (int ci = 0; ci < 128; ++ci) {
      const float e = __expf(sc[row][part * 128 + ci] - rm);
      sc[row][part * 128 + ci] = e;
      s += e;
    }
    red[part][row] = s;
  }
  __syncthreads();
  if (part == 0)
    rowsum[row] = red[0][row] + red[1][row] + red[2][row] + red[3][row];
  __syncthreads();
  {
    const float inv = 1.0f / rowsum[row];
    for (int ci = 0; ci < 128; ++ci) sc[row][part * 128 + ci] *= inv;
  }
  __syncthreads();

  // phase 3: out(16q x 32hd) = attn(16x512) · V(512x32); wave w owns hd tile w
  {
    v8f acc = vz;
    const int hd0 = w * 16;
    const float* vrow = vp + (size_t)(hd0 + lm) * TS_ + 2 * lh;
    const float* srow = &sc[lm][2 * lh];
    #pragma unroll 8
    for (int kk = 0; kk < S_; kk += 4) {
      const float2 av = *(const float2*)(srow + kk);  // LDS A frag (b64)
      const float2 bv = *(const float2*)(vrow + kk);  // global B frag (b64)
      v2f a;  a.x  = av.x; a.y  = av.y;
      v2f bb; bb.x = bv.x; bb.y = bv.y;
      acc = wmma4(a, bb, acc);
    }
    float* vo = vals + head_off + (size_t)(hd0 + lm) * TS_ + qBase + 8 * lh;
    *(float4*)(vo)     = make_float4(acc[0], acc[1], acc[2], acc[3]);
    *(float4*)(vo + 4) = make_float4(acc[4], acc[5], acc[6], acc[7]);
  }
}

// ---------------- host orchestration ----------------
extern "C" void kernel_launch(void* const* d_in, const int* in_sizes, int n_in,
                              void* d_out, int out_size, void* d_ws, size_t ws_size,
                              hipStream_t stream) {
  (void)in_sizes; (void)n_in; (void)out_size; (void)ws_size;
  const float* decoder = (const float*)d_in[0];
  const float* encoder = (const float*)d_in[1];
  const float* pos     = (const float*)d_in[2];
  const float* coord   = (const float*)d_in[3];
  const unsigned char* mask = (const unsigned char*)d_in[4];
  const float* Wq = (const float*)d_in[5];  const float* bq = (const float*)d_in[6];
  const float* Wk = (const float*)d_in[7];  const float* bk = (const float*)d_in[8];
  const float* Wv = (const float*)d_in[9];  const float* bv = (const float*)d_in[10];
  const float* Wo = (const float*)d_in[11]; const float* bo = (const float*)d_in[12];
  const float* W1 = (const float*)d_in[13]; const float* b1 = (const float*)d_in[14];
  const float* W2 = (const float*)d_in[15]; const float* b2 = (const float*)d_in[16];
  const float* ln_g = (const float*)d_in[17];
  const float* ln_b = (const float*)d_in[18];

  float* dec  = (float*)d_out;          // residual stream, updated in place
  float* src2 = (float*)d_ws;           // 5 x 16MB workspace slots
  float* qb   = src2 + NTOT_;
  float* kb   = src2 + 2 * NTOT_;
  float* vb   = src2 + 3 * NTOT_;
  float* valb = src2 + 4 * NTOT_;
  float* hb   = qb;                     // FFN hidden chunk reuses q slot

  hipMemcpyAsync(dec, decoder, NTOT_ * sizeof(float),
                 hipMemcpyDeviceToDevice, stream);

  const dim3 gProj(TS_ / 64, C_ / 64, B_);
  const dim3 gAttn(S_ / 16, H_ * T_, B_);

  for (int i = 0; i < L_; ++i) {
    const float* Wq_i = Wq + (size_t)i * C_ * C_;   const float* bq_i = bq + i * C_;
    const float* Wk_i = Wk + (size_t)i * C_ * C_;   const float* bk_i = bk + i * C_;
    const float* Wv_i = Wv + (size_t)i * C_ * C_;   const float* bv_i = bv + i * C_;
    const float* Wo_i = Wo + (size_t)i * C_ * C_;   const float* bo_i = bo + i * C_;
    const float* W1_i = W1 + (size_t)i * DFF_ * C_; const float* b1_i = b1 + i * DFF_;
    const float* W2_i = W2 + (size_t)i * C_ * DFF_; const float* b2_i = b2 + i * C_;
    const float* g0 = ln_g + (size_t)(i * 3 + 0) * C_; const float* e0 = ln_b + (size_t)(i * 3 + 0) * C_;
    const float* g1 = ln_g + (size_t)(i * 3 + 1) * C_; const float* e1 = ln_b + (size_t)(i * 3 + 1) * C_;
    const float* g2 = ln_g + (size_t)(i * 3 + 2) * C_; const float* e2 = ln_b + (size_t)(i * 3 + 2) * C_;

    // --- block 1: self-attention, Q/K = LN(dec)+pos, V = LN(dec) ---
    ln_kernel<<<B_ * TS_, C_, 0, stream>>>(dec, src2, g0, e0);
    gemm_kernel<C_, TS_, true,  false, false><<<gProj, 128, 0, stream>>>(Wq_i, bq_i, src2, pos, nullptr, qb, C_, 0, TS_, 0);
    gemm_kernel<C_, TS_, true,  false, false><<<gProj, 128, 0, stream>>>(Wk_i, bk_i, src2, pos, nullptr, kb, C_, 0, TS_, 0);
    gemm_kernel<C_, TS_, false, false, false><<<gProj, 128, 0, stream>>>(Wv_i, bv_i, src2, nullptr, nullptr, vb, C_, 0, TS_, 0);
    attn_kernel<<<gAttn, 64, 0, stream>>>(qb, kb, vb, coord, mask, valb);
    gemm_kernel<C_, TS_, false, true,  false><<<gProj, 128, 0, stream>>>(Wo_i, bo_i, valb, nullptr, dec, dec, C_, 0, TS_, 0);

    // --- block 2: Q/K = encoder+pos, V = LN(dec) (same attn weights) ---
    ln_kernel<<<B_ * TS_, C_, 0, stream>>>(dec, src2, g1, e1);
    gemm_kernel<C_, TS_, true,  false, false><<<gProj, 128, 0, stream>>>(Wq_i, bq_i, encoder, pos, nullptr, qb, C_, 0, TS_, 0);
    gemm_kernel<C_, TS_, true,  false, false><<<gProj, 128, 0, stream>>>(Wk_i, bk_i, encoder, pos, nullptr, kb, C_, 0, TS_, 0);
    gemm_kernel<C_, TS_, false, false, false><<<gProj, 128, 0, stream>>>(Wv_i, bv_i, src2, nullptr, nullptr, vb, C_, 0, TS_, 0);
    attn_kernel<<<gAttn, 64, 0, stream>>>(qb, kb, vb, coord, mask, valb);
    gemm_kernel<C_, TS_, false, true,  false><<<gProj, 128, 0, stream>>>(Wo_i, bo_i, valb, nullptr, dec, dec, C_, 0, TS_, 0);

    // --- FFN, N chunked by 2048 so hidden buffer fits one 16MB slot ---
    ln_kernel<<<B_ * TS_, C_, 0, stream>>>(dec, src2, g2, e2);
    for (int cch = 0; cch < 4; ++cch) {
      const int n0 = cch * 2048;
      gemm_kernel<C_, TS_, false, false, true><<<dim3(2048 / 64, DFF_ / 64, B_), 128, 0, stream>>>(
          W1_i, b1_i, src2, nullptr, nullptr, hb, DFF_, n0, 2048, 0);
      gemm_kernel<DFF_, 2048, false, true, false><<<dim3(2048 / 64, C_ / 64, B_), 128, 0, stream>>>(
          W2_i, b2_i, hb, nullptr, dec, dec, C_, 0, TS_, n0);
    }
  }
}